// sumNN_57982058496157
// MI455X (gfx1250) — compile-verified
//
#include <hip/hip_runtime.h>
#include <math.h>

typedef __attribute__((ext_vector_type(2))) float v2f;
typedef __attribute__((ext_vector_type(8))) float v8f;

#define B_TOT  8192
#define L_SEQ  50
#define D_EMB  128
#define C_HID  256
#define R_OUT  7
#define ROWS   32      // batch rows per block (two WMMA M-tiles -> B fragments reused 2x)
#define STRIDE 260     // padded LDS row stride in floats (260 % 64 == 4 -> conflict-free)

// Masked bag sum for one (row, side). Lane owns cols lane*4..lane*4+3 of D=128.
// Tokens+masks fetched with 2 coalesced per-lane loads, broadcast via __shfl (wave32).
__device__ __forceinline__ void bag_sum(const int* __restrict__ toks,
                                        const int* __restrict__ msks,
                                        const float* __restrict__ emb,
                                        int row_g, int lane, float4& acc)
{
    const int base = row_g * L_SEQ;
    int t0 = toks[base + lane];                  // l = 0..31
    int m0 = msks[base + lane];
    int t1 = -1, m1 = 0;
    if (lane < L_SEQ - 32) {                     // l = 32..49
        t1 = toks[base + 32 + lane];
        m1 = msks[base + 32 + lane];
    }
    const int e0 = m0 ? t0 : -1;                 // fold mask into token
    const int e1 = m1 ? t1 : -1;

    #pragma unroll 4
    for (int l = 0; l < 32; ++l) {
        const int t = __shfl(e0, l, 32);         // wave-uniform broadcast
        if (t >= 0) {                            // uniform branch: skip masked gathers
            const float4 e = *(const float4*)(emb + (size_t)t * D_EMB + lane * 4);
            acc.x += e.x; acc.y += e.y; acc.z += e.z; acc.w += e.w;
        }
    }
    #pragma unroll 3
    for (int l = 0; l < L_SEQ - 32; ++l) {
        const int t = __shfl(e1, l, 32);
        if (t >= 0) {
            const float4 e = *(const float4*)(emb + (size_t)t * D_EMB + lane * 4);
            acc.x += e.x; acc.y += e.y; acc.z += e.z; acc.w += e.w;
        }
    }
}

__launch_bounds__(512, 1)
__global__ void fused_embbag_mlp_kernel(
    const int*   __restrict__ lt,  const int*   __restrict__ rt,
    const int*   __restrict__ lm,  const int*   __restrict__ rm,
    const float* __restrict__ emb,
    const float* __restrict__ Wc,  const float* __restrict__ bc,
    const float* __restrict__ Ws,  const float* __restrict__ bs,
    float* __restrict__ out)
{
    __shared__ float smem[ROWS * STRIDE];   // 33.3 KB: concat tile, then reused for h tile

    const int tid  = threadIdx.x;
    const int wave = tid >> 5;              // 0..15
    const int lane = tid & 31;

    // ---------------- Phase 1: bag sums (L2-resident gather) ----------------
    // wave owns local rows {wave, wave+16}, both sides each.
    #pragma unroll
    for (int rr = 0; rr < 2; ++rr) {
        const int row_l = wave + rr * 16;
        const int row_g = blockIdx.x * ROWS + row_l;
        float4 accL = make_float4(0.f, 0.f, 0.f, 0.f);
        float4 accR = make_float4(0.f, 0.f, 0.f, 0.f);
        bag_sum(lt, lm, emb, row_g, lane, accL);
        bag_sum(rt, rm, emb, row_g, lane, accR);
        *(float4*)(&smem[row_l * STRIDE + lane * 4])       = accL;  // cols [0,128)
        *(float4*)(&smem[row_l * STRIDE + 128 + lane * 4]) = accR;  // cols [128,256)
    }

    __syncthreads();

    // -------- Phase 2: h = concat @ Wc^T  via V_WMMA_F32_16X16X4_F32 --------
    // wave owns output cols [n_base, n_base+16) for BOTH row tiles (M=0..15 and 16..31);
    // the two WMMAs per k-step share one B fragment -> W_cpr traffic halved.
    const int    n_base = wave * 16;
    const int    mrow   = lane & 15;           // A: M index / B: N index
    const int    kofs   = (lane >> 4) * 2;     // upper half-wave handles K+2,K+3
    const float* wrow   = Wc + (size_t)(n_base + mrow) * C_HID;

    v8f c0 = {0.f, 0.f, 0.f, 0.f, 0.f, 0.f, 0.f, 0.f};
    v8f c1 = {0.f, 0.f, 0.f, 0.f, 0.f, 0.f, 0.f, 0.f};
    #pragma unroll 8
    for (int k = 0; k < C_HID; k += 4) {
        const v2f b  = *(const v2f*)(&wrow[k + kofs]);                         // B[k.., n]
        const v2f a0 = *(const v2f*)(&smem[mrow * STRIDE + k + kofs]);         // rows 0..15
        const v2f a1 = *(const v2f*)(&smem[(mrow + 16) * STRIDE + k + kofs]);  // rows 16..31
        c0 = __builtin_amdgcn_wmma_f32_16x16x4_f32(false, a0, false, b, (short)0, c0, false, false);
        c1 = __builtin_amdgcn_wmma_f32_16x16x4_f32(false, a1, false, b, (short)0, c1, false, false);
    }

    // -------- Phase 3: bias + LeakyReLU, write h tiles back to LDS --------
    const float bias = bc[n_base + mrow];
    float hv0[8], hv1[8];
    #pragma unroll
    for (int i = 0; i < 8; ++i) {
        const float v0 = c0[i] + bias;
        const float v1 = c1[i] + bias;
        hv0[i] = (v0 >= 0.f) ? v0 : 0.01f * v0;
        hv1[i] = (v1 >= 0.f) ? v1 : 0.01f * v1;
    }
    __syncthreads();   // all waves done reading concat tile
    #pragma unroll
    for (int i = 0; i < 8; ++i) {
        const int m = (lane < 16) ? i : (i + 8);   // C/D layout: upper lanes hold M+8
        smem[m * STRIDE + n_base + mrow]        = hv0[i];
        smem[(m + 16) * STRIDE + n_base + mrow] = hv1[i];
    }
    __syncthreads();

    // -------- Phase 4: logits = h @ Ws^T + bs, then log_softmax (R=7) --------
    #pragma unroll
    for (int rr = 0; rr < 2; ++rr) {
        const int row_l = wave + rr * 16;
        const int row_g = blockIdx.x * ROWS + row_l;

        float p[R_OUT];
        #pragma unroll
        for (int r = 0; r < R_OUT; ++r) p[r] = 0.f;

        const int kk = lane * 8;               // each lane covers 8 of 256 columns
        #pragma unroll
        for (int j = 0; j < 8; ++j) {
            const float h = smem[row_l * STRIDE + kk + j];
            #pragma unroll
            for (int r = 0; r < R_OUT; ++r)
                p[r] += h * Ws[r * C_HID + kk + j];
        }
        #pragma unroll
        for (int r = 0; r < R_OUT; ++r) {
            #pragma unroll
            for (int off = 16; off > 0; off >>= 1)
                p[r] += __shfl_xor(p[r], off, 32);
        }

        if (lane == 0) {
            float lg[R_OUT];
            float mx = -INFINITY;
            #pragma unroll
            for (int r = 0; r < R_OUT; ++r) {
                lg[r] = p[r] + bs[r];
                mx = fmaxf(mx, lg[r]);
            }
            float s = 0.f;
            #pragma unroll
            for (int r = 0; r < R_OUT; ++r) s += expf(lg[r] - mx);
            const float lse = mx + logf(s);
            #pragma unroll
            for (int r = 0; r < R_OUT; ++r)
                out[row_g * R_OUT + r] = lg[r] - lse;
        }
    }
}

extern "C" void kernel_launch(void* const* d_in, const int* in_sizes, int n_in,
                              void* d_out, int out_size, void* d_ws, size_t ws_size,
                              hipStream_t stream) {
    (void)in_sizes; (void)n_in; (void)out_size; (void)d_ws; (void)ws_size;
    const int*   lt  = (const int*)  d_in[0];   // left_tokens  [B,L]
    const int*   rt  = (const int*)  d_in[1];   // right_tokens [B,L]
    const int*   lm  = (const int*)  d_in[2];   // left_mask    [B,L]
    const int*   rm  = (const int*)  d_in[3];   // right_mask   [B,L]
    const float* emb = (const float*)d_in[4];   // emb          [V,D]
    const float* Wc  = (const float*)d_in[5];   // W_cpr        [C,2D]
    const float* bc  = (const float*)d_in[6];   // b_cpr        [C]
    const float* Ws  = (const float*)d_in[7];   // W_sm         [R,C]
    const float* bs  = (const float*)d_in[8];   // b_sm         [R]
    float* out = (float*)d_out;                 // [B,R]

    dim3 grid(B_TOT / ROWS);   // 256 blocks
    dim3 block(512);           // 16 waves (wave32)
    fused_embbag_mlp_kernel<<<grid, block, 0, stream>>>(
        lt, rt, lm, rm, emb, Wc, bc, Ws, bs, out);
}